// CollatzGNN_13924283973778
// MI455X (gfx1250) — compile-verified
//
#include <hip/hip_runtime.h>

// ---------------------------------------------------------------------------
// 3-layer GCN for MI455X (gfx1250).
//  - Dense transforms via V_WMMA_F32_16X16X4_F32 (fp32 WMMA, 16x16 tile/wave)
//  - Edge aggregation via float4 gathers + f32 global atomics (L2-resident)
//  - Fused self-loop + bias + ReLU epilogue
// ---------------------------------------------------------------------------

typedef __attribute__((ext_vector_type(2))) float v2f;
typedef __attribute__((ext_vector_type(8))) float v8f;

#define GCN_N 100000           // N_NODES (multiple of 16: 6250 row tiles)

// ---- degree / normalization -----------------------------------------------

__global__ void k_init_deg(float* __restrict__ deg) {
    int i = blockIdx.x * blockDim.x + threadIdx.x;
    if (i < GCN_N) deg[i] = 1.0f;          // self-loop contributes 1
}

__global__ void k_deg_accum(const int* __restrict__ dst, int E,
                            float* __restrict__ deg) {
    int e = blockIdx.x * blockDim.x + threadIdx.x;
    if (e < E) atomicAdd(&deg[dst[e]], 1.0f);
}

__global__ void k_deg_rsqrt(float* __restrict__ deg) {
    int i = blockIdx.x * blockDim.x + threadIdx.x;
    if (i < GCN_N) deg[i] = rsqrtf(deg[i]);   // deg >= 1 always (self-loops)
}

__global__ void k_zero(float* __restrict__ p, int n) {
    int i = blockIdx.x * blockDim.x + threadIdx.x;
    if (i < n) p[i] = 0.0f;
}

// ---- dense transform: T[N,NO] = X[N,KI] @ W[KI,NO] via fp32 WMMA ----------
// One wave computes one 16x16 tile of T, accumulating K in chunks of 4 with
// V_WMMA_F32_16X16X4_F32.
//   A (16x4 f32, 2 VGPR): lanes 0-15 hold (M=lane, K={0,1}); lanes 16-31 hold
//                         (M=lane-16, K={2,3}).
//   B (4x16 f32, 2 VGPR): lanes 0-15 hold (N=lane, K={0,1}); lanes 16-31 hold
//                         (N=lane-16, K={2,3}).
//   D (16x16 f32, 8 VGPR): VGPR j -> (M = j + 8*(lane>=16), N = lane&15).
template <int KI, int NO>
__global__ void k_gemm_wmma(const float* __restrict__ X,
                            const float* __restrict__ W,
                            float* __restrict__ T) {
    const int wave     = (blockIdx.x * blockDim.x + threadIdx.x) >> 5;
    const int lane     = threadIdx.x & 31;
    const int nColTile = NO / 16;
    const int rowTile  = wave / nColTile;
    const int colTile  = wave - rowTile * nColTile;
    if (rowTile >= GCN_N / 16) return;      // uniform per wave: EXEC stays all-1

    const int row0 = rowTile * 16;
    const int col0 = colTile * 16;
    const int m    = lane & 15;             // row (A) / col (B,D) within tile
    const int hi   = lane >> 4;             // 0: K pair {0,1}; 1: K pair {2,3}

    v8f acc = {};
    constexpr int KCHUNKS = (KI + 3) / 4;
#pragma unroll
    for (int kk = 0; kk < KCHUNKS; ++kk) {
        const int kb = kk * 4 + hi * 2;
        v2f a, b;
        a.x = (kb     < KI) ? X[(row0 + m) * KI + kb    ] : 0.0f;
        a.y = (kb + 1 < KI) ? X[(row0 + m) * KI + kb + 1] : 0.0f;
        b.x = (kb     < KI) ? W[(kb    ) * NO + col0 + m] : 0.0f;
        b.y = (kb + 1 < KI) ? W[(kb + 1) * NO + col0 + m] : 0.0f;
        // 8 args: (neg_a, A, neg_b, B, c_mod, C, reuse_a, reuse_b)
        acc = __builtin_amdgcn_wmma_f32_16x16x4_f32(
            false, a, false, b, (short)0, acc, false, false);
    }

#pragma unroll
    for (int j = 0; j < 8; ++j)
        T[(row0 + j + hi * 8) * NO + col0 + m] = acc[j];
}

// ---- edge scatter: agg[dst] += T[src] * dinv[src]*dinv[dst] ---------------
template <int F>
__global__ void k_scatter(const int* __restrict__ src,
                          const int* __restrict__ dst, int E,
                          const float* __restrict__ dinv,
                          const float* __restrict__ T,
                          float* __restrict__ agg) {
    int e = blockIdx.x * blockDim.x + threadIdx.x;
    if (e >= E) return;
    const int   s    = src[e];
    const int   d    = dst[e];
    const float norm = dinv[s] * dinv[d];
    const float4* tp = (const float4*)(T + (size_t)s * F);
    float*        ap = agg + (size_t)d * F;
#pragma unroll
    for (int c = 0; c < F / 4; ++c) {
        float4 v = tp[c];
        atomicAdd(ap + 4 * c + 0, v.x * norm);
        atomicAdd(ap + 4 * c + 1, v.y * norm);
        atomicAdd(ap + 4 * c + 2, v.z * norm);
        atomicAdd(ap + 4 * c + 3, v.w * norm);
    }
}

// ---- epilogue: agg += self-loop + bias, optional ReLU (in place) ----------
template <int F, bool RELU>
__global__ void k_finish(const float* __restrict__ T,
                         const float* __restrict__ dinv,
                         const float* __restrict__ bias,
                         float* __restrict__ agg) {
    int i = blockIdx.x * blockDim.x + threadIdx.x;
    if (i >= GCN_N * F) return;
    const int   node = i / F;
    const int   f    = i - node * F;
    const float di   = dinv[node];
    float v = agg[i] + T[i] * di * di + bias[f];
    agg[i] = RELU ? fmaxf(v, 0.0f) : v;
}

// ---------------------------------------------------------------------------

extern "C" void kernel_launch(void* const* d_in, const int* in_sizes, int n_in,
                              void* d_out, int out_size, void* d_ws, size_t ws_size,
                              hipStream_t stream) {
    const float* x  = (const float*)d_in[0];
    const int*   ei = (const int*)d_in[1];
    const float* W1 = (const float*)d_in[2];
    const float* b1 = (const float*)d_in[3];
    const float* W2 = (const float*)d_in[4];
    const float* b2 = (const float*)d_in[5];
    const float* W3 = (const float*)d_in[6];
    const float* b3 = (const float*)d_in[7];
    float* out = (float*)d_out;

    const int E = in_sizes[1] / 2;          // edge_index is [2, E] flattened
    const int* src = ei;
    const int* dst = ei + E;

    const int N = GCN_N;

    // workspace layout (floats)
    float* ws   = (float*)d_ws;
    float* dinv = ws;                 // N       (deg -> rsqrt in place)
    float* t1   = dinv + N;           // N*16
    float* h1   = t1 + (size_t)N * 16;// N*16
    float* t2   = h1 + (size_t)N * 16;// N*32
    float* h2   = t2 + (size_t)N * 32;// N*32
    float* t3   = h2 + (size_t)N * 32;// N*16

    const int B = 256;
    auto blk = [](int n, int b) { return (n + b - 1) / b; };

    // --- degree normalization ---
    k_init_deg<<<blk(N, B), B, 0, stream>>>(dinv);
    k_deg_accum<<<blk(E, B), B, 0, stream>>>(dst, E, dinv);
    k_deg_rsqrt<<<blk(N, B), B, 0, stream>>>(dinv);

    // --- layer 1: 2 -> 16, ReLU ---
    {
        const int waves = (N / 16) * 1;
        k_gemm_wmma<2, 16><<<blk(waves * 32, B), B, 0, stream>>>(x, W1, t1);
        k_zero<<<blk(N * 16, B), B, 0, stream>>>(h1, N * 16);
        k_scatter<16><<<blk(E, B), B, 0, stream>>>(src, dst, E, dinv, t1, h1);
        k_finish<16, true><<<blk(N * 16, B), B, 0, stream>>>(t1, dinv, b1, h1);
    }

    // --- layer 2: 16 -> 32, ReLU ---
    {
        const int waves = (N / 16) * 2;
        k_gemm_wmma<16, 32><<<blk(waves * 32, B), B, 0, stream>>>(h1, W2, t2);
        k_zero<<<blk(N * 32, B), B, 0, stream>>>(h2, N * 32);
        k_scatter<32><<<blk(E, B), B, 0, stream>>>(src, dst, E, dinv, t2, h2);
        k_finish<32, true><<<blk(N * 32, B), B, 0, stream>>>(t2, dinv, b2, h2);
    }

    // --- layer 3: 32 -> 16, no ReLU, scatter straight into d_out ---
    {
        const int waves = (N / 16) * 1;
        k_gemm_wmma<32, 16><<<blk(waves * 32, B), B, 0, stream>>>(h2, W3, t3);
        k_zero<<<blk(N * 16, B), B, 0, stream>>>(out, N * 16);
        k_scatter<16><<<blk(E, B), B, 0, stream>>>(src, dst, E, dinv, t3, out);
        k_finish<16, false><<<blk(N * 16, B), B, 0, stream>>>(t3, dinv, b3, out);
    }
}